// KroneckerLinear_82282983457228
// MI455X (gfx1250) — compile-verified
//
#include <hip/hip_runtime.h>

// KroneckerLinear on gfx1250: Y[tok] = sum_s A_s @ X @ B_s^T + bias, all fp32,
// via native V_WMMA_F32_16X16X4_F32 (fp32 in/out matches reference numerics).
//
// GEMM order chosen for K-contiguity of every WMMA operand in LDS:
//   Stage 1: U_s = X . B_s^T   (M=i, N=p, K=j)  A-op = X rows, B-op = B rows
//   Stage 2: Y   = sum_s A_s . U_s  (M=o, N=p, K=i)  A-op = A rows, B-op = Ut rows
// U is spilled transposed (Ut[p][i]) so stage-2's B-operand is row-contiguous;
// the transposed spill merges into 2x ds_store_b128 per 16x16 tile.
//
// Roofline: 34.4 GFLOP over 512 MB HBM traffic (x in + y out) = 67 FLOP/B;
// HBM floor ~22us @ 23.3 TB/s, f32 WMMA compute in the same ballpark.

#if defined(__HIP_DEVICE_COMPILE__)
#if !__has_builtin(__builtin_amdgcn_wmma_f32_16x16x4_f32)
#error "device pass: missing __builtin_amdgcn_wmma_f32_16x16x4_f32"
#endif
#endif

typedef __attribute__((ext_vector_type(2))) float v2f;
typedef __attribute__((ext_vector_type(4))) float v4f;
typedef __attribute__((ext_vector_type(8))) float v8f;

#define IN1 64
#define IN2 64
#define OUT1 64
#define OUT2 64
#define NSUM 2
#define TOKENS (4 * 4096)
#define TOK_PER_WG 8
#define THREADS 256
#define LSTRIDE 68 // padded LDS row stride (floats): 16B-aligned rows, breaks
                   // the stride-64 bank aliasing between WMMA lane halves

// LDS layout (float offsets)
#define OFF_A 0                              // A[s][o][i] row-major, NSUM*64*LSTRIDE
#define OFF_B (OFF_A + NSUM * 64 * LSTRIDE)  // B[s][p][j] row-major (untransposed)
#define OFF_X (OFF_B + NSUM * 64 * LSTRIDE)  // X[i][j], 64*LSTRIDE
#define OFF_UT (OFF_X + 64 * LSTRIDE)        // Ut[s][p][i], NSUM*64*LSTRIDE
#define OFF_BIAST (OFF_UT + NSUM * 64 * LSTRIDE) // biasT[p][o], 64*LSTRIDE
#define LDS_FLOATS (OFF_BIAST + 64 * LSTRIDE)    // 34816 floats = 139264 B

__global__ __launch_bounds__(THREADS) void kron_wmma_f32(
    const float* __restrict__ x, const float* __restrict__ A,
    const float* __restrict__ B, const float* __restrict__ bias,
    float* __restrict__ y) {
  extern __shared__ float lds[];
  float* sA = lds + OFF_A;
  float* sB = lds + OFF_B;
  float* sX = lds + OFF_X;
  float* sUt = lds + OFF_UT;
  float* sBiasT = lds + OFF_BIAST;

  const int tid = threadIdx.x;
  const int lane = tid & 31;
  const int wave = tid >> 5;
  const int h = lane >> 4;  // lane half (selects K pair / M half per ISA layout)
  const int lm = lane & 15; // lane within half (M row for A-op, N col for B/D)

  // ---- One-time per-workgroup preload of A, B, biasT into LDS ----
  for (int idx = tid; idx < NSUM * 64 * 64; idx += THREADS) {
    const int s = idx >> 12;
    const int r = (idx >> 6) & 63; // A: row o | B: row p
    const int c = idx & 63;        // A: col i | B: col j
    sA[(s * 64 + r) * LSTRIDE + c] = A[idx];
    sB[(s * 64 + r) * LSTRIDE + c] = B[idx]; // keep row-major: rows are K-contig
  }
  for (int idx = tid; idx < 64 * 64; idx += THREADS) {
    const int o = idx >> 6, p = idx & 63;
    sBiasT[p * LSTRIDE + o] = bias[idx]; // transposed: matches D-tile row order
  }
  __syncthreads();

  for (int it = 0; it < TOK_PER_WG; ++it) {
    const long long token = (long long)blockIdx.x * TOK_PER_WG + it;
    const float* xt = x + token * (IN1 * IN2);
    float* yt = y + token * (OUT1 * OUT2);

    // ---- Load X (64x64 f32, 16 KB) into LDS, coalesced float4 ----
    for (int q = tid; q < (IN1 * IN2) / 4; q += THREADS) {
      const int r = q >> 4;  // row i
      const int c4 = q & 15; // float4 within row
      *(v4f*)(sX + r * LSTRIDE + c4 * 4) = *(const v4f*)(xt + q * 4);
    }
    __syncthreads();

    // ---- Stage 1: U_s = X . B_s^T  (32 tiles across 8 waves) ----
#pragma unroll
    for (int q = 0; q < 4; ++q) {
      const int job = wave * 4 + q;  // 0..31
      const int s = job >> 4;        // summand
      const int ti = (job >> 2) & 3; // i-tile (M)
      const int tp = job & 3;        // p-tile (N)
      // A-op: X[i = ti*16+lm][k..k+1]; B-op: B[s][p = tp*16+lm][k..k+1]
      const float* a_base = sX + (ti * 16 + lm) * LSTRIDE;
      const float* b_base = sB + (s * 64 + tp * 16 + lm) * LSTRIDE;
      v8f acc = {};
#pragma unroll
      for (int k0 = 0; k0 < 64; k0 += 4) {
        const int ka = k0 + 2 * h;
        const v2f av = *(const v2f*)(a_base + ka);
        const v2f bv = *(const v2f*)(b_base + ka);
        acc = __builtin_amdgcn_wmma_f32_16x16x4_f32(false, av, false, bv,
                                                    (short)0, acc, false, false);
      }
      // Transposed spill: Ut[p][i] ; D rows are i = v + 8h (v consecutive)
      float* u_base = sUt + (s * 64 + tp * 16 + lm) * LSTRIDE + ti * 16 + 8 * h;
      v4f lo = {acc[0], acc[1], acc[2], acc[3]};
      v4f hi = {acc[4], acc[5], acc[6], acc[7]};
      *(v4f*)(u_base) = lo;     // ds_store_b128
      *(v4f*)(u_base + 4) = hi; // ds_store_b128
    }
    __syncthreads();

    // ---- Stage 2: Y = sum_s A_s . U_s + bias (16 tiles / 8 waves) ----
#pragma unroll
    for (int q = 0; q < 2; ++q) {
      const int job = wave * 2 + q; // 0..15
      const int to = job >> 2;      // o-tile (M)
      const int tp = job & 3;       // p-tile (N)
      v8f acc = {};
#pragma unroll
      for (int s = 0; s < NSUM; ++s) {
        // A-op: A[s][o = to*16+lm][k..k+1]; B-op: Ut[s][p = tp*16+lm][k..k+1]
        const float* a_base = sA + (s * 64 + to * 16 + lm) * LSTRIDE;
        const float* b_base = sUt + (s * 64 + tp * 16 + lm) * LSTRIDE;
#pragma unroll
        for (int k0 = 0; k0 < 64; k0 += 4) {
          const int ka = k0 + 2 * h;
          const v2f av = *(const v2f*)(a_base + ka);
          const v2f bv = *(const v2f*)(b_base + ka);
          acc = __builtin_amdgcn_wmma_f32_16x16x4_f32(
              false, av, false, bv, (short)0, acc, false, false);
        }
      }
      // Bias (transposed layout -> 2x b128 LDS reads), then store to global.
      const float* bias_base =
          sBiasT + (tp * 16 + lm) * LSTRIDE + to * 16 + 8 * h;
      const v4f blo = *(const v4f*)(bias_base);
      const v4f bhi = *(const v4f*)(bias_base + 4);
      float* y_base = yt + (to * 16 + 8 * h) * OUT2 + tp * 16 + lm;
#pragma unroll
      for (int v = 0; v < 4; ++v) y_base[v * OUT2] = acc[v] + blo[v];
#pragma unroll
      for (int v = 0; v < 4; ++v) y_base[(v + 4) * OUT2] = acc[v + 4] + bhi[v];
    }
    __syncthreads(); // protect X/Ut before next token overwrites them
  }
}

extern "C" void kernel_launch(void* const* d_in, const int* in_sizes, int n_in,
                              void* d_out, int out_size, void* d_ws,
                              size_t ws_size, hipStream_t stream) {
  (void)in_sizes;
  (void)n_in;
  (void)d_ws;
  (void)ws_size;
  (void)out_size;
  const float* x = (const float*)d_in[0];
  const float* A = (const float*)d_in[1];
  const float* B = (const float*)d_in[2];
  const float* bias = (const float*)d_in[3];
  float* y = (float*)d_out;

  const size_t lds_bytes = (size_t)LDS_FLOATS * sizeof(float); // ~136 KB (<320 KB/WGP)
  const int grid = TOKENS / TOK_PER_WG;                        // 2048 workgroups
  kron_wmma_f32<<<grid, THREADS, lds_bytes, stream>>>(x, A, B, bias, y);
}